// GNN_87205015978673
// MI455X (gfx1250) — compile-verified
//
#include <hip/hip_runtime.h>

typedef float v2f __attribute__((ext_vector_type(2)));
typedef float v8f __attribute__((ext_vector_type(8)));

#define EP_DUAL  0
#define EP_RELU  1
#define EP_STATS 2

// ---------------------------------------------------------------------------
// Generic fp32 WMMA GEMM:  D[M x NB] = A[M x KDIM] @ B[KDIM x NB] + bias
// block = NB threads (NB/32 waves); block computes 16 rows x NB cols.
// Each wave computes two 16x16 tiles via V_WMMA_F32_16X16X4_F32, K chained.
// Epilogues: DUAL (store to D and D2), RELU, STATS (store + col sum/sumsq).
// ---------------------------------------------------------------------------
template <int KDIM, int NB, int EPI>
__global__ __launch_bounds__(NB) void wmma_gemm_kernel(
    const float* __restrict__ A, const float* __restrict__ B,
    const float* __restrict__ bias, float* __restrict__ D,
    float* __restrict__ D2, float* __restrict__ stats)
{
    constexpr int LDA = KDIM + 4;                 // pad -> conflict-free LDS
    constexpr int K4  = KDIM / 4;
    __shared__ float As[16 * LDA];

    const int tid = threadIdx.x;
    const int rowbase = blockIdx.x * 16;

    // cooperative b128 load of the 16 x KDIM A tile
    for (int idx = tid; idx < 16 * K4; idx += NB) {
        const int m = idx / K4, k4 = idx - m * K4;
        *(float4*)&As[m * LDA + k4 * 4] =
            *(const float4*)&A[(size_t)(rowbase + m) * KDIM + k4 * 4];
    }
    __syncthreads();

    const int wave = tid >> 5;
    const int lane = tid & 31;
    const int hi   = lane >> 4;     // half-wave selects K pair / row group
    const int ln   = lane & 15;     // M for A-frag, N for B-frag / C cols
    const int col0 = wave * 32;
    const int col1 = col0 + 16;

    v8f acc0 = {};
    v8f acc1 = {};

#pragma unroll
    for (int kk = 0; kk < KDIM / 4; ++kk) {
        const int k0 = kk * 4 + 2 * hi;
        v2f a, b0, b1;
        a.x  = As[ln * LDA + k0];
        a.y  = As[ln * LDA + k0 + 1];
        b0.x = B[(size_t)k0 * NB + col0 + ln];
        b0.y = B[(size_t)(k0 + 1) * NB + col0 + ln];
        b1.x = B[(size_t)k0 * NB + col1 + ln];
        b1.y = B[(size_t)(k0 + 1) * NB + col1 + ln];
        acc0 = __builtin_amdgcn_wmma_f32_16x16x4_f32(false, a, false, b0,
                                                     (short)0, acc0, false, false);
        acc1 = __builtin_amdgcn_wmma_f32_16x16x4_f32(false, a, false, b1,
                                                     (short)0, acc1, false, false);
    }

    const float bb0 = bias[col0 + ln];
    const float bb1 = bias[col1 + ln];
    float s0 = 0.f, q0 = 0.f, s1 = 0.f, q1 = 0.f;

#pragma unroll
    for (int r = 0; r < 8; ++r) {
        const int m = rowbase + r + 8 * hi;       // C/D layout: M = r + 8*hi
        float v0 = acc0[r] + bb0;
        float v1 = acc1[r] + bb1;
        if (EPI == EP_RELU) { v0 = fmaxf(v0, 0.f); v1 = fmaxf(v1, 0.f); }
        D[(size_t)m * NB + col0 + ln] = v0;
        D[(size_t)m * NB + col1 + ln] = v1;
        if (EPI == EP_DUAL) {
            D2[(size_t)m * NB + col0 + ln] = v0;
            D2[(size_t)m * NB + col1 + ln] = v1;
        }
        if (EPI == EP_STATS) {
            s0 += v0; q0 += v0 * v0;
            s1 += v1; q1 += v1 * v1;
        }
    }
    if (EPI == EP_STATS) {
        atomicAdd(&stats[col0 + ln], s0);
        atomicAdd(&stats[NB + col0 + ln], q0);
        atomicAdd(&stats[col1 + ln], s1);
        atomicAdd(&stats[NB + col1 + ln], q1);
    }
}

// ---------------------------------------------------------------------------
// Message passing: for each edge e, msg = relu(h[src] + edge_attr[e] @ W + b)
// scattered with atomicAdd into z[dst]. W (16x256) lives in LDS; edge_attr is
// recomputed per layer (cheap) instead of materializing ea[E,256] (307 MB).
// 256 threads = 4 edge slots x 64 threads; each thread owns 4 channels
// (b128 gather, b128 LDS weight reads; atomics must stay b32).
// ---------------------------------------------------------------------------
__global__ __launch_bounds__(256) void scatter_kernel(
    const float* __restrict__ h, const int* __restrict__ ei,
    const float* __restrict__ eattr, const float* __restrict__ ew,
    const float* __restrict__ eb, float* __restrict__ z, int E)
{
    __shared__ float Ws[16 * 256];
    __shared__ float sEb[256];
    __shared__ float sAttr[16][16];
    __shared__ int   sSrc[16];
    __shared__ int   sDst[16];

    const int t = threadIdx.x;
    {
        const float4* w4 = (const float4*)ew;
        float4* s4 = (float4*)Ws;
        for (int j = t; j < 16 * 64; j += 256) s4[j] = w4[j];
        sEb[t] = eb[t];
    }

    const int slot = t >> 6;          // which of 4 edges this thread works on
    const int c4   = (t & 63) * 4;    // channel group [c4, c4+4)

    const int ebase = blockIdx.x * 64;
    for (int chunk = 0; chunk < 4; ++chunk) {
        __syncthreads();
        const int e0 = ebase + chunk * 16;
        {
            const int i = t >> 4, k = t & 15;
            const int e = e0 + i;
            sAttr[i][k] = (e < E) ? eattr[(size_t)e * 16 + k] : 0.f;
        }
        if (t < 16) {
            const int e = e0 + t;
            sSrc[t] = (e < E) ? ei[e] : 0;
        } else if (t < 32) {
            const int e = e0 + t - 16;
            sDst[t - 16] = (e < E) ? ei[E + e] : 0;
        }
        __syncthreads();

        for (int i = 0; i < 4; ++i) {
            const int li = i * 4 + slot;          // edge within chunk
            const int e  = e0 + li;
            if (e >= E) continue;                 // uniform per wave
            float4 v = *(const float4*)&sEb[c4];
#pragma unroll
            for (int k = 0; k < 16; ++k) {
                const float  a = sAttr[li][k];
                const float4 w = *(const float4*)&Ws[k * 256 + c4];
                v.x += a * w.x; v.y += a * w.y;
                v.z += a * w.z; v.w += a * w.w;
            }
            const float4 hv = *(const float4*)&h[(size_t)sSrc[li] * 256 + c4];
            v.x = fmaxf(v.x + hv.x, 0.f);
            v.y = fmaxf(v.y + hv.y, 0.f);
            v.z = fmaxf(v.z + hv.z, 0.f);
            v.w = fmaxf(v.w + hv.w, 0.f);
            float* zp = &z[(size_t)sDst[li] * 256 + c4];
            atomicAdd(zp + 0, v.x);
            atomicAdd(zp + 1, v.y);
            atomicAdd(zp + 2, v.z);
            atomicAdd(zp + 3, v.w);
        }
    }
}

__global__ void zero_kernel(float* __restrict__ p, int n)
{
    const int i = blockIdx.x * blockDim.x + threadIdx.x;
    if (i < n) p[i] = 0.f;
}

__global__ __launch_bounds__(256) void bn_finalize_kernel(
    const float* __restrict__ stats, const float* __restrict__ gamma,
    const float* __restrict__ beta, float* __restrict__ scsh, float invN)
{
    const int c = threadIdx.x;
    const float mean = stats[c] * invN;
    const float var  = stats[256 + c] * invN - mean * mean;
    const float sc   = gamma[c] * rsqrtf(var + 1e-5f);
    scsh[c]       = sc;
    scsh[256 + c] = beta[c] - mean * sc;
}

// h = relu(z*scale + shift); also re-seeds z (next layer's agg accumulator).
// Fully float4: i indexes groups of 4 consecutive channels.
__global__ __launch_bounds__(256) void bn_relu_kernel(
    float4* __restrict__ z, float4* __restrict__ h,
    const float* __restrict__ scsh, int total4)
{
    const int i = blockIdx.x * blockDim.x + threadIdx.x;
    if (i >= total4) return;
    const int c4 = (i & 63) * 4;
    const float4 sc = *(const float4*)&scsh[c4];
    const float4 sh = *(const float4*)&scsh[256 + c4];
    float4 v = z[i];
    v.x = fmaxf(fmaf(v.x, sc.x, sh.x), 0.f);
    v.y = fmaxf(fmaf(v.y, sc.y, sh.y), 0.f);
    v.z = fmaxf(fmaf(v.z, sc.z, sh.z), 0.f);
    v.w = fmaxf(fmaf(v.w, sc.w, sh.w), 0.f);
    z[i] = v;
    h[i] = v;
}

// 4 nodes per block, 64 threads x float4 channels each.
__global__ __launch_bounds__(256) void pool_kernel(
    const float* __restrict__ h, const int* __restrict__ batch,
    float* __restrict__ g, int N)
{
    const int t = threadIdx.x;
    const int i = blockIdx.x * 4 + (t >> 6);
    if (i >= N) return;
    const int c4 = (t & 63) * 4;
    const int b  = batch[i];
    const float4 v = *(const float4*)&h[(size_t)i * 256 + c4];
    float* gp = &g[(size_t)b * 256 + c4];
    atomicAdd(gp + 0, v.x);
    atomicAdd(gp + 1, v.y);
    atomicAdd(gp + 2, v.z);
    atomicAdd(gp + 3, v.w);
}

// out[128,12] = hidden[128,128] @ head_w2[128,12] + head_b2 (tiny, 1 block)
__global__ __launch_bounds__(256) void head2_kernel(
    const float* __restrict__ hidden, const float* __restrict__ w2,
    const float* __restrict__ b2, float* __restrict__ out)
{
    for (int idx = threadIdx.x; idx < 128 * 12; idx += 256) {
        const int gi = idx / 12, o = idx - gi * 12;
        float s = b2[o];
#pragma unroll 8
        for (int k = 0; k < 128; ++k) s += hidden[gi * 128 + k] * w2[k * 12 + o];
        out[idx] = s;
    }
}

extern "C" void kernel_launch(void* const* d_in, const int* in_sizes, int n_in,
                              void* d_out, int out_size, void* d_ws, size_t ws_size,
                              hipStream_t stream)
{
    const float* x      = (const float*)d_in[0];
    const int*   ei     = (const int*)  d_in[1];
    const int*   batch  = (const int*)  d_in[2];
    const float* eattr  = (const float*)d_in[3];
    const float* node_w = (const float*)d_in[4];
    const float* node_b = (const float*)d_in[5];
    const float* edge_w = (const float*)d_in[6];
    const float* edge_b = (const float*)d_in[7];
    const float* mlp_w1 = (const float*)d_in[8];
    const float* mlp_b1 = (const float*)d_in[9];
    const float* mlp_w2 = (const float*)d_in[10];
    const float* mlp_b2 = (const float*)d_in[11];
    const float* bn_g   = (const float*)d_in[12];
    const float* bn_b   = (const float*)d_in[13];
    const float* hw1    = (const float*)d_in[14];
    const float* hb1    = (const float*)d_in[15];
    const float* hw2    = (const float*)d_in[16];
    const float* hb2    = (const float*)d_in[17];
    float* out = (float*)d_out;

    constexpr int N = 50000, E = 300000, G = 128, HID = 256, L = 4;

    char* ws = (char*)d_ws;
    size_t off = 0;
    float* H     = (float*)(ws + off); off += (size_t)N * HID * 4;   // node feats
    float* Z     = (float*)(ws + off); off += (size_t)N * HID * 4;   // agg / z2
    float* T     = (float*)(ws + off); off += (size_t)N * HID * 4;   // mlp hidden
    float* stats = (float*)(ws + off); off += 512 * 4;               // col sum/sumsq
    float* scsh  = (float*)(ws + off); off += 512 * 4;               // BN scale/shift
    float* Gp    = (float*)(ws + off); off += (size_t)G * HID * 4;   // pooled
    float* Hd    = (float*)(ws + off); off += (size_t)G * 128 * 4;   // head hidden

    // node encoder: H = Z = x @ node_w + node_b   (Z doubles as agg init)
    wmma_gemm_kernel<32, 256, EP_DUAL><<<N / 16, 256, 0, stream>>>(
        x, node_w, node_b, H, Z, nullptr);

    for (int l = 0; l < L; ++l) {
        // z = h + sum_{e: dst=v} relu(h[src] + edge_mlp(edge_attr))
        scatter_kernel<<<(E + 63) / 64, 256, 0, stream>>>(
            H, ei, eattr, edge_w, edge_b, Z, E);
        // t = relu(z @ w1 + b1)
        wmma_gemm_kernel<256, 256, EP_RELU><<<N / 16, 256, 0, stream>>>(
            Z, mlp_w1 + (size_t)l * HID * HID, mlp_b1 + l * HID, T, nullptr, nullptr);
        // z2 = t @ w2 + b2  (+ fused column sum / sumsq for BatchNorm)
        zero_kernel<<<2, 256, 0, stream>>>(stats, 512);
        wmma_gemm_kernel<256, 256, EP_STATS><<<N / 16, 256, 0, stream>>>(
            T, mlp_w2 + (size_t)l * HID * HID, mlp_b2 + l * HID, Z, nullptr, stats);
        bn_finalize_kernel<<<1, 256, 0, stream>>>(
            stats, bn_g + l * HID, bn_b + l * HID, scsh, 1.0f / (float)N);
        // h = relu(BN(z2)); z re-seeded with h for next layer's aggregation
        bn_relu_kernel<<<(N * HID / 4 + 255) / 256, 256, 0, stream>>>(
            (float4*)Z, (float4*)H, scsh, N * HID / 4);
    }

    // global_add_pool
    zero_kernel<<<(G * HID) / 256, 256, 0, stream>>>(Gp, G * HID);
    pool_kernel<<<(N + 3) / 4, 256, 0, stream>>>(H, batch, Gp, N);

    // head: hidden = relu(g @ head_w1 + b1)  [128x128], then tiny final GEMM
    wmma_gemm_kernel<256, 128, EP_RELU><<<G / 16, 128, 0, stream>>>(
        Gp, hw1, hb1, Hd, nullptr, nullptr);
    head2_kernel<<<1, 256, 0, stream>>>(Hd, hw2, hb2, out);
}